// DeChunkLayer_13331578486925
// MI455X (gfx1250) — compile-verified
//
#include <hip/hip_runtime.h>

// ---------------------------------------------------------------------------
// DeChunk EMA scan for MI455X (gfx1250, wave32).
//
// z_t = (1-p_t) z_{t-1} + p_t x_t  over (L=16384, D=1024), then expand to
// L_FULL=32768 rows by boundary runs. Memory-bound (~320MB -> ~14us @23.3TB/s).
// The scan is blocked into 16-step sub-chunks and evaluated as
//   Z(16 x 16ch) = T(16x16) @ X(16 x 16ch) + c (outer) z_in
// using V_WMMA_F32_16X16X4_F32 (4 K-slices + 1 carry WMMA per tile).
// ---------------------------------------------------------------------------

#define D_DIM   1024
#define LCOMP   16384
#define LFULL   32768
#define CHUNK   128
#define NCHUNK  (LCOMP / CHUNK)   // 128
#define EPS_P   1e-4f

typedef __attribute__((ext_vector_type(2))) float v2f;
typedef __attribute__((ext_vector_type(8))) float v8f;

// --- Kernel 1: clip p, a = 1-p, per-chunk decay products ---------------------
__global__ void prep_kernel(const float* __restrict__ p_sel,
                            float* __restrict__ p_arr,
                            float* __restrict__ a_arr,
                            float* __restrict__ A_c) {
  const int tid = threadIdx.x;
  for (int i = tid; i < LCOMP; i += 1024) {
    float p = p_sel[i];
    p = fminf(fmaxf(p, EPS_P), 1.0f - EPS_P);
    p_arr[i] = p;
    a_arr[i] = 1.0f - p;
  }
  __syncthreads();
  if (tid < NCHUNK) {
    float prod = 1.0f;
    const float* a = a_arr + tid * CHUNK;
    for (int i = 0; i < CHUNK; ++i) prod *= a[i];
    A_c[tid] = prod;
  }
}

// --- Kernel 2: start[t] = position of t-th boundary (block scan of b) --------
__global__ void startpos_kernel(const int* __restrict__ b,
                                int* __restrict__ start) {
  __shared__ int sums[1024];
  const int tid = threadIdx.x;
  const int PER = LFULL / 1024;  // 32
  const int base = tid * PER;
  int loc[PER];
  int s = 0;
#pragma unroll
  for (int i = 0; i < PER; ++i) { s += b[base + i]; loc[i] = s; }
  sums[tid] = s;
  __syncthreads();
  for (int off = 1; off < 1024; off <<= 1) {
    int add = (tid >= off) ? sums[tid - off] : 0;
    __syncthreads();
    sums[tid] += add;
    __syncthreads();
  }
  const int excl = sums[tid] - s;
#pragma unroll
  for (int i = 0; i < PER; ++i) {
    if (b[base + i]) start[excl + loc[i] - 1] = base + i;
  }
  if (tid == 0) start[LCOMP] = LFULL;
}

// --- Kernel 4: serial scan over chunk aggregates -----------------------------
__global__ void state_scan_kernel(const float* __restrict__ A_c,
                                  const float* __restrict__ Bc,
                                  float* __restrict__ z_in) {
  const int d = threadIdx.x;  // 1024 channels
  float z = 0.0f;
  for (int c = 0; c < NCHUNK; ++c) {
    z_in[c * D_DIM + d] = z;
    z = A_c[c] * z + Bc[c * D_DIM + d];
  }
}

// --- Kernels 3 & 5: WMMA tile scan over one chunk ----------------------------
// PASS 0: z_in = 0, emit chunk-final state to Bc.
// PASS 1: load z_in, scatter each compressed row t to out[start[t]..start[t+1]).
template <int PASS>
__global__ void chunk_scan_kernel(const float* __restrict__ x,
                                  const float* __restrict__ p_arr,
                                  const float* __restrict__ a_arr,
                                  const float* __restrict__ z_in,
                                  const int* __restrict__ start,
                                  float* __restrict__ Bc,
                                  float* __restrict__ out) {
  __shared__ float sh_p[CHUNK];
  __shared__ float sh_a[CHUNK];
  const int c    = blockIdx.x;
  const int tid  = threadIdx.x;
  const int lane = tid & 31;
  const int wave = tid >> 5;
  const int m    = lane & 15;
  const bool hi  = lane >= 16;
  const int dch  = blockIdx.y * 128 + wave * 16 + m;  // this lane's channel
  const int tbase = c * CHUNK;

  if (tid < CHUNK)            sh_p[tid]         = p_arr[tbase + tid];
  else if (tid < 2 * CHUNK)   sh_a[tid - CHUNK] = a_arr[tbase + tid - CHUNK];
  __syncthreads();

  float z_cur = (PASS == 1) ? z_in[c * D_DIM + dch] : 0.0f;
  const int loCol = hi ? 2 : 0;  // A-matrix K columns held by this lane half

  for (int sc = 0; sc < CHUNK / 16; ++sc) {
    const int t0l = sc * 16;
    const int t0g = tbase + t0l;

    // Build triangular T row m: T[m][s] = p_s * prod_{u=s+1..m} a_u
    // (stable backward recurrence; r ends as cvec[m] = prod_{u<=m} a_u).
    v2f A[4] = {};
    float r = 1.0f;
#pragma unroll
    for (int s = 15; s >= 0; --s) {
      const float ps = sh_p[t0l + s];
      const float as = sh_a[t0l + s];
      const bool act = (s <= m);
      const float Tms = act ? r * ps : 0.0f;
      const int j   = s >> 2;
      const int rem = s & 3;
      A[j].x = (rem == loCol)     ? Tms : A[j].x;
      A[j].y = (rem == loCol + 1) ? Tms : A[j].y;
      r = act ? r * as : r;
    }

    // B tiles: B[j][k][n] = x[t0+4j+k][d0+n]; VGPR0 = K{0,2}, VGPR1 = K{1,3}.
    v2f B[4];
#pragma unroll
    for (int j = 0; j < 4; ++j) {
      const int k0 = 4 * j + (hi ? 2 : 0);
      const float* px = x + (size_t)(t0g + k0) * D_DIM + dch;
      B[j].x = px[0];
      B[j].y = px[D_DIM];
    }

    // Carry term as outer product: column 0 of A = cvec, row 0 of B = z_in.
    v2f cA, cB;
    cA.x = hi ? 0.0f : r;      cA.y = 0.0f;
    cB.x = hi ? 0.0f : z_cur;  cB.y = 0.0f;

    v8f acc = {};
    acc = __builtin_amdgcn_wmma_f32_16x16x4_f32(false, cA, false, cB,
                                                (short)0, acc, false, false);
#pragma unroll
    for (int j = 0; j < 4; ++j)
      acc = __builtin_amdgcn_wmma_f32_16x16x4_f32(false, A[j], false, B[j],
                                                  (short)0, acc, false, false);

    // State at sub-chunk end = row M=15 (lanes 16..31, VGPR 7); broadcast.
    z_cur = __shfl(acc[7], 16 + m, 32);

    if (PASS == 1) {
      // Scatter row t to its run [start[t], start[t+1]) in the full sequence.
#pragma unroll
      for (int rI = 0; rI < 8; ++rI) {
        const int tg = t0g + (hi ? 8 : 0) + rI;
        const int s0 = start[tg];
        const int s1 = start[tg + 1];
        const float v = acc[rI];
        for (int tf = s0; tf < s1; ++tf)
          out[(size_t)tf * D_DIM + dch] = v;
      }
    }
  }

  if (PASS == 0 && !hi) Bc[c * D_DIM + dch] = z_cur;
}

// ---------------------------------------------------------------------------
extern "C" void kernel_launch(void* const* d_in, const int* in_sizes, int n_in,
                              void* d_out, int out_size, void* d_ws, size_t ws_size,
                              hipStream_t stream) {
  const float* x     = (const float*)d_in[0];   // (1, 16384, 1024) f32
  const float* p_sel = (const float*)d_in[1];   // (16384,) f32
  const int*   b     = (const int*)d_in[2];     // (1, 32768) i32
  float* out = (float*)d_out;                   // (1, 32768, 1024) f32

  float* ws    = (float*)d_ws;
  float* p_arr = ws;                              // 16384
  float* a_arr = p_arr + LCOMP;                   // 16384
  float* A_c   = a_arr + LCOMP;                   // 128 (pad to 256)
  float* Bc    = A_c + 256;                       // NCHUNK*D = 131072
  float* z_in  = Bc + NCHUNK * D_DIM;             // 131072
  int*   start = (int*)(z_in + NCHUNK * D_DIM);   // 16385 ints
  // total workspace ~1.25 MB

  prep_kernel<<<1, 1024, 0, stream>>>(p_sel, p_arr, a_arr, A_c);
  startpos_kernel<<<1, 1024, 0, stream>>>(b, start);

  dim3 grid(NCHUNK, D_DIM / 128);  // 128 chunks x 8 channel groups
  chunk_scan_kernel<0><<<grid, 256, 0, stream>>>(x, p_arr, a_arr, z_in, start, Bc, out);
  state_scan_kernel<<<1, 1024, 0, stream>>>(A_c, Bc, z_in);
  chunk_scan_kernel<1><<<grid, 256, 0, stream>>>(x, p_arr, a_arr, z_in, start, Bc, out);
}